// GATModel_83623013253775
// MI455X (gfx1250) — compile-verified
//
#include <hip/hip_runtime.h>
#include <math.h>

// GAT (2-layer, PyG-style) for MI455X / gfx1250.
// GEMMs via V_WMMA_F32_16X16X4_F32 (fp32 matches reference numerics; GEMM is
// not the bottleneck). Edge softmax + aggregation via global atomics
// (bandwidth-bound: ~2.3 GB traffic => ~0.1 ms at 23.3 TB/s HBM).
// Message pass vectorized float4 (global_load_b128 + 4x global_atomic_add_f32).

#define IN_CH 128
#define HID   64
#define HEADS 4
#define NEG_SLOPE 0.2f

typedef float v2f __attribute__((ext_vector_type(2)));
typedef float v8f __attribute__((ext_vector_type(8)));

// ---- order-preserving float <-> uint encoding (for atomic float max) ----
__device__ __forceinline__ unsigned f2ord(float f) {
  unsigned u = __float_as_uint(f);
  return (u & 0x80000000u) ? ~u : (u | 0x80000000u);
}
__device__ __forceinline__ float ord2f(unsigned u) {
  return (u & 0x80000000u) ? __uint_as_float(u & 0x7fffffffu)
                           : __uint_as_float(~u);
}
#define NEG_INF_ORD 0x007FFFFFu   // f2ord(-inf)

__device__ __forceinline__ float elu_f(float x) {
  return x > 0.0f ? x : (expf(x) - 1.0f);
}

// ------------------------- init fills -------------------------
__global__ void fill_f32(float* p, float v, int n) {
  int i = blockIdx.x * blockDim.x + threadIdx.x;
  if (i < n) p[i] = v;
}
__global__ void fill_u32(unsigned* p, unsigned v, int n) {
  int i = blockIdx.x * blockDim.x + threadIdx.x;
  if (i < n) p[i] = v;
}

// ------------------------- WMMA GEMM (fp32) -------------------------
// C[M,Nc] = A[M,K] @ B[K,Nc], row-major. One wave per 16x16 tile.
// A frag (16x4 f32): lane<16 -> M=lane, K={k0,k0+1}; lane>=16 -> M=lane-16, K={k0+2,k0+3}.
// B frag (4x16 f32): mirrored: v0 = B[k0+2*hi][n0+lr], v1 = B[k0+2*hi+1][n0+lr].
// C/D (16x16 f32): row = v + 8*hi, col = lr.
template <int K>
__global__ void wmma_gemm_f32(const float* __restrict__ A,
                              const float* __restrict__ B,
                              float* __restrict__ C, int M, int Nc) {
  int wave = (int)((blockIdx.x * blockDim.x + threadIdx.x) >> 5);
  int ntn  = Nc >> 4;
  int total = (M >> 4) * ntn;       // M, Nc are multiples of 16 here
  if (wave >= total) return;        // uniform per wave: EXEC stays all-ones
  int tm = wave / ntn, tn = wave % ntn;
  int lane = (int)(threadIdx.x & 31);
  int lr = lane & 15, hi = lane >> 4;

  const float* arow = A + (size_t)(tm * 16 + lr) * K;
  const float* bcol = B + tn * 16 + lr;
  v8f acc = {};
#pragma unroll 8
  for (int k0 = 0; k0 < K; k0 += 4) {
    int ka = k0 + hi * 2;
    v2f a; a.x = arow[ka];               a.y = arow[ka + 1];
    v2f b; b.x = bcol[(size_t)ka * Nc];  b.y = bcol[(size_t)(ka + 1) * Nc];
    acc = __builtin_amdgcn_wmma_f32_16x16x4_f32(
        /*neg_a=*/false, a, /*neg_b=*/false, b,
        /*c_mod=*/(short)0, acc, /*reuse_a=*/false, /*reuse_b=*/false);
  }
  float* crow = C + tn * 16 + lr;
  int rbase = tm * 16 + hi * 8;
#pragma unroll
  for (int v = 0; v < 8; ++v) crow[(size_t)(rbase + v) * Nc] = acc[v];
}

// ---------------- per-node attention halves: es/ed [n,H] ----------------
template <int H>
__global__ void attn_coef(const float* __restrict__ h,
                          const float* __restrict__ a_src,
                          const float* __restrict__ a_dst,
                          float* __restrict__ es, float* __restrict__ ed, int n) {
  int idx = blockIdx.x * blockDim.x + threadIdx.x;   // over n*H
  if (idx >= n * H) return;
  int i = idx / H, hh = idx % H;
  const float* hp = h + (size_t)i * (H * HID) + hh * HID;
  const float* as = a_src + hh * HID;
  const float* ad = a_dst + hh * HID;
  float s = 0.f, d = 0.f;
#pragma unroll 8
  for (int c = 0; c < HID; ++c) { float v = hp[c]; s += v * as[c]; d += v * ad[c]; }
  es[idx] = s; ed[idx] = d;
}

// ------------- edge pass A: leaky-relu logits + segment max -------------
template <int H>
__global__ void edge_logits(const int* __restrict__ ei,
                            const float* __restrict__ es,
                            const float* __restrict__ ed,
                            float* __restrict__ logit,
                            unsigned* __restrict__ mEnc, int E_, int N_) {
  int idx = blockIdx.x * blockDim.x + threadIdx.x;   // over (E_+N_)*H
  if (idx >= (E_ + N_) * H) return;
  int e = idx / H, hh = idx % H;
  int s = (e < E_) ? ei[e]      : (e - E_);
  int d = (e < E_) ? ei[E_ + e] : (e - E_);
  float l = es[s * H + hh] + ed[d * H + hh];
  l = (l > 0.0f) ? l : NEG_SLOPE * l;
  logit[idx] = l;
  atomicMax(&mEnc[d * H + hh], f2ord(l));
}

// ------------- edge pass B: exp(logit - max), segment sum -------------
template <int H>
__global__ void edge_expsum(const int* __restrict__ ei,
                            const float* __restrict__ logit,
                            const unsigned* __restrict__ mEnc,
                            float* __restrict__ ssum, int E_, int N_) {
  int idx = blockIdx.x * blockDim.x + threadIdx.x;
  if (idx >= (E_ + N_) * H) return;
  int e = idx / H, hh = idx % H;
  int d = (e < E_) ? ei[E_ + e] : (e - E_);
  float m = ord2f(mEnc[d * H + hh]);
  atomicAdd(&ssum[d * H + hh], expf(logit[idx] - m));
}

// ------ edge pass C: att = exp/sum; scatter-add messages (HBM-bound) ------
// One thread per 4 channels: float4 gather (global_load_b128) + 4 atomic adds;
// attention coefficient computed once per 4 channels.
template <int H>
__global__ void edge_message4(const int* __restrict__ ei,
                              const float* __restrict__ h,
                              const float* __restrict__ logit,
                              const unsigned* __restrict__ mEnc,
                              const float* __restrict__ ssum,
                              float* __restrict__ agg, int E_, int N_) {
  const int LDH = H * HID;
  const int LD4 = LDH / 4;
  int idx = blockIdx.x * blockDim.x + threadIdx.x;   // over (E_+N_)*LD4 (<2^31)
  if (idx >= (E_ + N_) * LD4) return;
  int e = idx / LD4;
  int t = (idx % LD4) * 4;                            // channel base (16B aligned)
  int hh = t >> 6;                                    // HID == 64
  int s = (e < E_) ? ei[e]      : (e - E_);
  int d = (e < E_) ? ei[E_ + e] : (e - E_);
  float m = ord2f(mEnc[d * H + hh]);
  float a = expf(logit[e * H + hh] - m) / (ssum[d * H + hh] + 1e-16f);
  const float4 hv = *reinterpret_cast<const float4*>(h + (size_t)s * LDH + t);
  float* ap = agg + (size_t)d * LDH + t;
  atomicAdd(ap + 0, hv.x * a);
  atomicAdd(ap + 1, hv.y * a);
  atomicAdd(ap + 2, hv.z * a);
  atomicAdd(ap + 3, hv.w * a);
}

// ------------------------- bias + ELU -------------------------
template <int LD>
__global__ void bias_elu(const float* __restrict__ agg,
                         const float* __restrict__ bias,
                         float* __restrict__ out, int total) {
  int idx = blockIdx.x * blockDim.x + threadIdx.x;
  if (idx >= total) return;
  out[idx] = elu_f(agg[idx] + bias[idx % LD]);
}

// ------------- final projection: one wave32 per node, coalesced -------------
__global__ void out_proj(const float* __restrict__ h,
                         const float* __restrict__ Wo,
                         const float* __restrict__ bo,
                         float* __restrict__ out, int n) {
  int wid  = (int)((blockIdx.x * blockDim.x + threadIdx.x) >> 5);  // node
  int lane = (int)(threadIdx.x & 31);
  if (wid >= n) return;
  const float* hp = h + (size_t)wid * HID;
  float s = hp[lane] * Wo[lane] + hp[lane + 32] * Wo[lane + 32];
#pragma unroll
  for (int o = 16; o > 0; o >>= 1) s += __shfl_xor(s, o, 32);
  if (lane == 0) out[wid] = s + bo[0];
}

#define CDIV(a, b) (((a) + (b) - 1) / (b))

extern "C" void kernel_launch(void* const* d_in, const int* in_sizes, int n_in,
                              void* d_out, int out_size, void* d_ws, size_t ws_size,
                              hipStream_t stream) {
  const float* x   = (const float*)d_in[0];
  const float* W1  = (const float*)d_in[1];
  const float* as1 = (const float*)d_in[2];
  const float* ad1 = (const float*)d_in[3];
  const float* b1  = (const float*)d_in[4];
  const float* W2  = (const float*)d_in[5];
  const float* as2 = (const float*)d_in[6];
  const float* ad2 = (const float*)d_in[7];
  const float* b2  = (const float*)d_in[8];
  const float* Wo  = (const float*)d_in[9];
  const float* bo  = (const float*)d_in[10];
  const int*   ei  = (const int*)d_in[11];

  const int n  = in_sizes[0]  / IN_CH;   // 50000 (multiple of 16)
  const int e  = in_sizes[11] / 2;       // 800000
  const int et = e + n;                  // edges incl. self-loops

  // ---- workspace layout (floats), ~149 MB total ----
  float* p = (float*)d_ws;
  float*    h1   = p;            p += (size_t)n * 256;   // X@W1, later elu(l1 out)
  float*    agg1 = p;            p += (size_t)n * 256;
  float*    h2   = p;            p += (size_t)n * 64;    // @W2, later elu(l2 out)
  float*    agg2 = p;            p += (size_t)n * 64;
  float*    es1  = p;            p += (size_t)n * 4;
  float*    ed1  = p;            p += (size_t)n * 4;
  unsigned* m1   = (unsigned*)p; p += (size_t)n * 4;
  float*    s1   = p;            p += (size_t)n * 4;
  float*    es2v = p;            p += n;
  float*    ed2v = p;            p += n;
  unsigned* m2   = (unsigned*)p; p += n;
  float*    s2   = p;            p += n;
  float*    lg1  = p;            p += (size_t)et * 4;
  float*    lg2  = p;            /* p += et; */

  const int B = 256;

  // ---- deterministic re-init of accumulators every call ----
  fill_f32<<<CDIV(n * 256, B), B, 0, stream>>>(agg1, 0.0f, n * 256);
  fill_f32<<<CDIV(n * 64,  B), B, 0, stream>>>(agg2, 0.0f, n * 64);
  fill_f32<<<CDIV(n * 4,   B), B, 0, stream>>>(s1,   0.0f, n * 4);
  fill_f32<<<CDIV(n,       B), B, 0, stream>>>(s2,   0.0f, n);
  fill_u32<<<CDIV(n * 4,   B), B, 0, stream>>>(m1, NEG_INF_ORD, n * 4);
  fill_u32<<<CDIV(n,       B), B, 0, stream>>>(m2, NEG_INF_ORD, n);

  // ---- layer 1 ----
  int tiles1 = (n / 16) * (256 / 16);
  wmma_gemm_f32<128><<<CDIV(tiles1, 8), B, 0, stream>>>(x, W1, h1, n, 256);
  attn_coef<HEADS><<<CDIV(n * HEADS, B), B, 0, stream>>>(h1, as1, ad1, es1, ed1, n);
  edge_logits<HEADS><<<CDIV(et * HEADS, B), B, 0, stream>>>(ei, es1, ed1, lg1, m1, e, n);
  edge_expsum<HEADS><<<CDIV(et * HEADS, B), B, 0, stream>>>(ei, lg1, m1, s1, e, n);
  edge_message4<HEADS><<<CDIV(et * 64, B), B, 0, stream>>>(ei, h1, lg1, m1, s1, agg1, e, n);
  bias_elu<256><<<CDIV(n * 256, B), B, 0, stream>>>(agg1, b1, h1, n * 256);

  // ---- layer 2 ----
  int tiles2 = (n / 16) * (64 / 16);
  wmma_gemm_f32<256><<<CDIV(tiles2, 8), B, 0, stream>>>(h1, W2, h2, n, 64);
  attn_coef<1><<<CDIV(n, B), B, 0, stream>>>(h2, as2, ad2, es2v, ed2v, n);
  edge_logits<1><<<CDIV(et, B), B, 0, stream>>>(ei, es2v, ed2v, lg2, m2, e, n);
  edge_expsum<1><<<CDIV(et, B), B, 0, stream>>>(ei, lg2, m2, s2, e, n);
  edge_message4<1><<<CDIV(et * 16, B), B, 0, stream>>>(ei, h2, lg2, m2, s2, agg2, e, n);
  bias_elu<64><<<CDIV(n * 64, B), B, 0, stream>>>(agg2, b2, h2, n * 64);

  // ---- output projection (wave per node, coalesced + shfl reduce) ----
  out_proj<<<CDIV(n * 32, B), B, 0, stream>>>(h2, Wo, bo, (float*)d_out, n);
}